// LaplacianLoss_80272938762697
// MI455X (gfx1250) — compile-verified
//
#include <hip/hip_runtime.h>

// LaplacianLoss on MI455X (gfx1250):
//   Y = L @ [X | 1]   (single pass over 256MB L, fp32 WMMA 16x16x4, split-K)
//   deg = Y[:,24]     (row sums == column sums, L symmetric)
//   loss = mean((Y[:, :24]/deg - X)^2)
//
// Memory-bound: 256MB of L at 23.3 TB/s ~= 11us floor. Split-K multiplies the
// wave count (up to 4096 waves) so enough loads are in flight to saturate HBM.
// L is streamed with non-temporal loads (used once); the 1MB RHS stays L2-hot.
// Wave-uniform values are readfirstlane'd into SGPRs so the K-loop uses scalar
// control flow (s_cmp/s_cbranch) instead of exec-mask looping.

#define NV    8192
#define BATCH 8
#define NCOL  32    // padded N: 24 data cols + 1 ones col + 7 zero cols
#define NDATA 24
#define LEPS  1e-6f

typedef __attribute__((ext_vector_type(2))) float v2f;
typedef __attribute__((ext_vector_type(8))) float v8f;

// ---------------------------------------------------------------------------
// Kernel 1: build padded transposed RHS  Xt[n][j],  n in [0,32), j in [0,8192)
//   n < 24 : Xt[n][j] = x[b=n/3][j][d=n%3] ; n == 24: 1.0f ; n > 24: 0.0f
// Transposed layout makes each WMMA B-fragment a contiguous float2 in K.
// ---------------------------------------------------------------------------
__global__ void __launch_bounds__(256)
build_xt_kernel(const float* __restrict__ x, float* __restrict__ xt) {
    int idx = blockIdx.x * 256 + threadIdx.x;   // 0 .. 32*8192-1
    int n = idx >> 13;                          // / 8192
    int j = idx & (NV - 1);
    float v;
    if (n < NDATA) {
        int b = n / 3;
        int d = n - 3 * b;
        v = x[(size_t)b * NV * 3 + (size_t)j * 3 + d];
    } else if (n == NDATA) {
        v = 1.0f;
    } else {
        v = 0.0f;
    }
    xt[(size_t)n * NV + j] = v;
}

// ---------------------------------------------------------------------------
// Kernel 2: split-K GEMM. Wave w handles M-tile (w / nsplit) over K-chunk
// (w % nsplit), accumulating a 16x32 partial with V_WMMA_F32_16X16X4_F32
// and writing it to its own slab ypart[chunk].
//
// f32 16x16x4 A fragment (2 VGPR/lane): lane holds row M = lane%16,
//   VGPR0 = K = (lane/16)*2, VGPR1 = K+1 -> contiguous float2 (NT load).
// B fragment mirrors A with N = lane%16:
//   contiguous float2 at Xt[n*NV + k]  (RT load, stays in L2).
// C/D: VGPR r holds (M = r + (lane/16)*8, N = lane%16).
// ---------------------------------------------------------------------------
__global__ void __launch_bounds__(128)
lap_gemm_kernel(const float* __restrict__ L, const float* __restrict__ xt,
                float* __restrict__ ypart, int nsplit, int kchunk) {
    const int lane = threadIdx.x & 31;

    // Wave-uniform tiling parameters -> force into SGPRs for scalar loop
    // control and SALU address math in the hot loop.
    const int w     = __builtin_amdgcn_readfirstlane(blockIdx.x * 4 + (threadIdx.x >> 5));
    const int tile  = __builtin_amdgcn_readfirstlane(w / nsplit);        // 0 .. 511
    const int chunk = __builtin_amdgcn_readfirstlane(w - tile * nsplit); // 0 .. nsplit-1
    const int rowBase = tile * 16;
    const int kBeg = __builtin_amdgcn_readfirstlane(chunk * kchunk);
    const int kEnd = __builtin_amdgcn_readfirstlane(kBeg + kchunk);

    const int m     = lane & 15;              // A row / B col within tile
    const int khalf = (lane >> 4) << 1;       // 0 or 2: lane's K offset

    const float* aPtr  = L  + (size_t)(rowBase + m) * NV + (size_t)(kBeg + khalf);
    const float* b0Ptr = xt + (size_t)m * NV + (size_t)(kBeg + khalf);        // N half 0
    const float* b1Ptr = xt + (size_t)(m + 16) * NV + (size_t)(kBeg + khalf); // N half 1

    v8f c0 = {};
    v8f c1 = {};

    const int kIter = kEnd - kBeg;            // scalar trip count
    #pragma unroll 1
    for (int k0 = 0; k0 < kIter; k0 += 16) {
        // speculative prefetch ~16 iterations ahead on the streaming L read
        __builtin_prefetch(aPtr + k0 + 256, 0, 0);
        #pragma unroll
        for (int s = 0; s < 4; ++s) {
            const int k = k0 + 4 * s;
            v2f a  = __builtin_nontemporal_load((const v2f*)(aPtr + k));
            v2f b0 = *(const v2f*)(b0Ptr + k);
            v2f b1 = *(const v2f*)(b1Ptr + k);
            c0 = __builtin_amdgcn_wmma_f32_16x16x4_f32(
                     false, a, false, b0, (short)0, c0, false, false);
            c1 = __builtin_amdgcn_wmma_f32_16x16x4_f32(
                     false, a, false, b1, (short)0, c1, false, false);
        }
    }

    // Write partial D tile: VGPR r -> (row = rowBase + r + (lane/16)*8, col = lane%16)
    float* yo = ypart + (size_t)chunk * NV * NCOL;   // chunk is an SGPR
    const int nlo = lane & 15;
    const int mhi = (lane >> 4) * 8;
    #pragma unroll
    for (int r = 0; r < 8; ++r) {
        const int row = rowBase + r + mhi;
        yo[(size_t)row * NCOL + nlo]      = c0[r];
        yo[(size_t)row * NCOL + 16 + nlo] = c1[r];
    }
}

// ---------------------------------------------------------------------------
// Kernel 3: per-row loss. Sums the nsplit partial slabs (deterministic order),
// normalizes by deg = Y[i,24] + EPS, accumulates squared diffs, block-reduces
// to 32 partial sums in workspace.
// ---------------------------------------------------------------------------
__global__ void __launch_bounds__(256)
lap_loss_partial_kernel(const float* __restrict__ ypart, const float* __restrict__ x,
                        float* __restrict__ partial, int nsplit) {
    __shared__ float smem[256];
    const int i = blockIdx.x * 256 + threadIdx.x;   // row index, 8192 total
    float acc = 0.0f;
    if (i < NV) {
        float deg = LEPS;
        for (int s = 0; s < nsplit; ++s)
            deg += ypart[(size_t)s * NV * NCOL + (size_t)i * NCOL + NDATA];
        #pragma unroll
        for (int n = 0; n < NDATA; ++n) {
            float ysum = 0.0f;
            for (int s = 0; s < nsplit; ++s)
                ysum += ypart[(size_t)s * NV * NCOL + (size_t)i * NCOL + n];
            const int b = n / 3;
            const int d = n - 3 * b;
            const float diff = ysum / deg - x[(size_t)b * NV * 3 + (size_t)i * 3 + d];
            acc += diff * diff;
        }
    }
    smem[threadIdx.x] = acc;
    __syncthreads();
    for (int off = 128; off > 0; off >>= 1) {
        if (threadIdx.x < off) smem[threadIdx.x] += smem[threadIdx.x + off];
        __syncthreads();
    }
    if (threadIdx.x == 0) partial[blockIdx.x] = smem[0];
}

// ---------------------------------------------------------------------------
// Kernel 4: final reduce of 32 partials -> scalar mean. Overwrites d_out
// deterministically every call.
// ---------------------------------------------------------------------------
__global__ void lap_loss_final_kernel(const float* __restrict__ partial,
                                      float* __restrict__ out) {
    if (threadIdx.x == 0) {
        float s = 0.0f;
        #pragma unroll
        for (int i = 0; i < 32; ++i) s += partial[i];
        out[0] = s * (1.0f / (float)(BATCH * NV * 3));
    }
}

// ---------------------------------------------------------------------------
// Launch. Inputs: d_in[0] = x (B,NV,3) f32 ; d_in[1] = laplacian (NV,NV) f32.
// Workspace: [Xt: 1MB][ypart: nsplit * 1MB][partials: 32 f32].
// nsplit picked from ws_size so the kernel adapts to available scratch.
// ---------------------------------------------------------------------------
extern "C" void kernel_launch(void* const* d_in, const int* in_sizes, int n_in,
                              void* d_out, int out_size, void* d_ws, size_t ws_size,
                              hipStream_t stream) {
    const float* x = (const float*)d_in[0];
    const float* L = (const float*)d_in[1];
    float* out = (float*)d_out;

    const size_t slab = (size_t)NV * NCOL;   // 262144 floats = 1MB
    int nsplit = 1;
    {
        const int cand[3] = {8, 4, 2};
        for (int c = 0; c < 3; ++c) {
            size_t need = (slab * (size_t)(1 + cand[c]) + 64) * sizeof(float);
            if (ws_size >= need) { nsplit = cand[c]; break; }
        }
    }
    const int kchunk = NV / nsplit;

    float* ws      = (float*)d_ws;
    float* xt      = ws;                               // 262144 floats
    float* ypart   = ws + slab;                        // nsplit * 262144 floats
    float* partial = ws + slab * (size_t)(1 + nsplit); // 32 floats

    build_xt_kernel<<<(NV * NCOL) / 256, 256, 0, stream>>>(x, xt);
    lap_gemm_kernel<<<(NV / 16) * nsplit / 4, 128, 0, stream>>>(L, xt, ypart, nsplit, kchunk);
    lap_loss_partial_kernel<<<NV / 256, 256, 0, stream>>>(ypart, x, partial, nsplit);
    lap_loss_final_kernel<<<1, 32, 0, stream>>>(partial, out);
}